// PrecRec_798863917464
// MI455X (gfx1250) — compile-verified
//
#include <hip/hip_runtime.h>
#include <hip/hip_bf16.h>
#include <stdint.h>

#define TILE   1024      // elements per block-tile (per array)
#define BLOCK  256       // 8 wave32 per block
#define EPT    4         // elements per thread per tile (one b128 per array)
#define DEPTH  4         // async pipeline stages (12 async ops in flight / wave)
#define MAXBLK 2048

// logit(i/11) for i = 1..10 (sigmoid(x) > i/11  <=>  x > logit(i/11))
__device__ __constant__ float kLogit[10] = {
    -2.3025850929940457f, -1.5040773967762742f, -0.9808292530117262f,
    -0.5596157879354227f, -0.1823215567939546f,
     0.1823215567939546f,  0.5596157879354227f,  0.9808292530117262f,
     1.5040773967762742f,  2.3025850929940457f};

// ws layout: [0..9]=tp_j, [10]=total_pos, [11..20]=fp_j, [21]=total_neg
__global__ void precrec_zero_ws(unsigned* ws) {
    if (threadIdx.x < 22) ws[threadIdx.x] = 0u;
}

__device__ __forceinline__ void upd(float x, int mk, int tg, unsigned cnt[11]) {
    // pos = mask & target, neg = mask & !target; pack pos in lo16, neg in hi16
    unsigned pos = (unsigned)((mk != 0) & (tg != 0));
    unsigned neg = (unsigned)((mk != 0) & (tg == 0));
    unsigned w   = pos | (neg << 16);
    cnt[10] += w;                       // totals
#pragma unroll
    for (int j = 0; j < 10; ++j)        // v_cmp + v_cndmask + v_add per threshold
        cnt[j] += (x > kLogit[j]) ? w : 0u;
}

__global__ __launch_bounds__(BLOCK) void precrec_main(
    const float* __restrict__ pred, const int* __restrict__ mask,
    const int* __restrict__ tgt, unsigned* __restrict__ ws,
    long long numTiles, long long N)
{
    __shared__ float sP[DEPTH][TILE];
    __shared__ int   sM[DEPTH][TILE];
    __shared__ int   sT[DEPTH][TILE];
    __shared__ unsigned sRed[BLOCK / 32][22];

    const int tid = threadIdx.x;
    unsigned cnt[11];
#pragma unroll
    for (int j = 0; j < 11; ++j) cnt[j] = 0u;

    // Async-copy one tile (3 x b128 per thread) into LDS ring slot `buf`.
    // GVS mode: mem = SGPR64 base + signed 32-bit VGPR byte offset (N*4 < 2^31 here).
    // LDS dest offset = low 32 bits of generic shared pointer. Non-temporal:
    // the stream is read-once (768MB >> 192MB L2), keep it out of the temporal set.
    auto issue = [&](int buf, long long tile) {
        const int      idx  = tid * EPT;
        const int      boff = (int)((tile * TILE + idx) * 4ll);
        const uint32_t lp   = (uint32_t)(uintptr_t)&sP[buf][idx];
        const uint32_t lm   = (uint32_t)(uintptr_t)&sM[buf][idx];
        const uint32_t lt   = (uint32_t)(uintptr_t)&sT[buf][idx];
        asm volatile("global_load_async_to_lds_b128 %0, %1, %2 th:TH_LOAD_NT"
                     :: "v"(lp), "v"(boff), "s"(pred) : "memory");
        asm volatile("global_load_async_to_lds_b128 %0, %1, %2 th:TH_LOAD_NT"
                     :: "v"(lm), "v"(boff), "s"(mask) : "memory");
        asm volatile("global_load_async_to_lds_b128 %0, %1, %2 th:TH_LOAD_NT"
                     :: "v"(lt), "v"(boff), "s"(tgt)  : "memory");
    };

    // Prologue: fill the ring with up to DEPTH tiles.
    long long t      = blockIdx.x;
    int       issued = 0;
#pragma unroll
    for (int s = 0; s < DEPTH; ++s) {
        const long long tt = t + (long long)s * gridDim.x;
        if (tt < numTiles) { issue(s, tt); ++issued; }
    }

    int buf = 0;
    for (; t < numTiles; t += gridDim.x) {
        // Oldest tile (in slot `buf`) is done when asynccnt <= 3*(issued-1);
        // async loads complete in order per wave.
        if      (issued >= 4) asm volatile("s_wait_asynccnt 0x9" ::: "memory");
        else if (issued == 3) asm volatile("s_wait_asynccnt 0x6" ::: "memory");
        else if (issued == 2) asm volatile("s_wait_asynccnt 0x3" ::: "memory");
        else                  asm volatile("s_wait_asynccnt 0x0" ::: "memory");

        const int    idx = tid * EPT;
        const float4 p = *reinterpret_cast<const float4*>(&sP[buf][idx]);
        const int4   m = *reinterpret_cast<const int4*>(&sM[buf][idx]);
        const int4   g = *reinterpret_cast<const int4*>(&sT[buf][idx]);
        upd(p.x, m.x, g.x, cnt);
        upd(p.y, m.y, g.y, cnt);
        upd(p.z, m.z, g.z, cnt);
        upd(p.w, m.w, g.w, cnt);
        --issued;

        // Recycle slot `buf` for tile t + DEPTH*gridDim. The pending ds_loads
        // above and a new async LDS-write are not ordered by HW: drain DScnt first.
        const long long tn = t + (long long)DEPTH * gridDim.x;
        if (tn < numTiles) {
            asm volatile("s_wait_dscnt 0x0" ::: "memory");
            issue(buf, tn);
            ++issued;
        }
        buf = (buf + 1) & (DEPTH - 1);
    }

    // tail (N not multiple of TILE): direct guarded loads
    const long long tailStart = numTiles * TILE;
    const long long gsz       = (long long)gridDim.x * BLOCK;
    for (long long i = tailStart + (long long)blockIdx.x * BLOCK + tid; i < N; i += gsz)
        upd(pred[i], mask[i], tgt[i], cnt);

    // wave32 reduction of packed counters (wave sums stay < 2^16 per half)
#pragma unroll
    for (int j = 0; j < 11; ++j) {
        unsigned v = cnt[j];
#pragma unroll
        for (int d = 16; d >= 1; d >>= 1) v += __shfl_xor(v, d, 32);
        cnt[j] = v;
    }
    const int wave = tid >> 5, lane = tid & 31;
    if (lane == 0) {
#pragma unroll
        for (int j = 0; j < 11; ++j) {
            sRed[wave][j]      = cnt[j] & 0xFFFFu;  // pos-class
            sRed[wave][11 + j] = cnt[j] >> 16;      // neg-class
        }
    }
    __syncthreads();
    if (tid < 22) {
        unsigned s = 0;
#pragma unroll
        for (int wv = 0; wv < BLOCK / 32; ++wv) s += sRed[wv][tid];
        atomicAdd(&ws[tid], s);
    }
}

// out (float, 40 elems): rows tp, fp, tn, fn over 10 thresholds
__global__ void precrec_finalize(const unsigned* __restrict__ ws,
                                 float* __restrict__ out) {
    const int tid = threadIdx.x;
    if (tid >= 40) return;
    const int j = tid % 10, r = tid / 10;
    const unsigned tp = ws[j], fp = ws[11 + j];
    const unsigned tpos = ws[10], tneg = ws[21];
    unsigned v = (r == 0) ? tp : (r == 1) ? fp : (r == 2) ? (tneg - fp) : (tpos - tp);
    out[tid] = (float)v;
}

extern "C" void kernel_launch(void* const* d_in, const int* in_sizes, int n_in,
                              void* d_out, int out_size, void* d_ws, size_t ws_size,
                              hipStream_t stream) {
    const float* pred = (const float*)d_in[0];
    const int*   mask = (const int*)d_in[1];
    const int*   tgt  = (const int*)d_in[2];
    unsigned*    ws   = (unsigned*)d_ws;
    const long long N        = (long long)in_sizes[0];
    const long long numTiles = N / TILE;
    long long nb = numTiles < 1 ? 1 : numTiles;
    if (nb > MAXBLK) nb = MAXBLK;
    const int blocks = (int)nb;

    precrec_zero_ws<<<1, 32, 0, stream>>>(ws);
    precrec_main<<<blocks, BLOCK, 0, stream>>>(pred, mask, tgt, ws, numTiles, N);
    precrec_finalize<<<1, 64, 0, stream>>>(ws, (float*)d_out);
}